// DiffGCN_46351287058748
// MI455X (gfx1250) — compile-verified
//
#include <hip/hip_runtime.h>
#include <hip/hip_bf16.h>
#include <math.h>

typedef __attribute__((ext_vector_type(2))) float v2f;
typedef __attribute__((ext_vector_type(8))) float v8f;

#define NNODES 50000
#define DEG    16
#define TSTEPS 4
#define CIN    128
#define HMLP   64
#define HGRU   128
#define WBC    704   /* 64 (hpre) + 4*64 (A_t) + 384 (GX) */

// ---------------------------------------------------------------------------
// Assemble Wbig (128 x 704):  [ W1_block0 | W1_block1..4 | Wx ]
// ---------------------------------------------------------------------------
__global__ void build_wbig(const float* __restrict__ W1,   // 640 x 64
                           const float* __restrict__ Wx,   // 128 x 384
                           float* __restrict__ Wbig) {
    int tid = blockIdx.x * blockDim.x + threadIdx.x;
    if (tid >= CIN * WBC) return;
    int r = tid / WBC;
    int c = tid % WBC;
    float v;
    if (c < 64) {
        v = W1[r * 64 + c];
    } else if (c < 320) {
        int t  = (c - 64) >> 6;
        int cc = (c - 64) & 63;
        v = W1[((1 + t) * CIN + r) * 64 + cc];
    } else {
        v = Wx[r * 384 + (c - 320)];
    }
    Wbig[tid] = v;
}

// ---------------------------------------------------------------------------
// Swizzle a K x Nc row-major B into the WMMA B-fragment layout:
//   SwB[ ((kb*(Nc/16)+cg)*32 + lane)*2 + p ] = B[ (kb*4 + (lane>>4)*2 + p)*Nc + cg*16 + (lane&15) ]
// so each wave's per-k-step B fragment for one 16-col group is a single
// contiguous 256B segment (one global_load_b64 per lane).
// ---------------------------------------------------------------------------
__global__ void swizzle_b(const float* __restrict__ B, float* __restrict__ SwB,
                          int K, int Nc) {
    int o = blockIdx.x * blockDim.x + threadIdx.x;
    if (o >= K * Nc) return;
    int p    = o & 1;
    int lane = (o >> 1) & 31;
    int rest = o >> 6;
    int ncg  = Nc >> 4;
    int cg   = rest % ncg;
    int kb   = rest / ncg;
    int row  = kb * 4 + ((lane >> 4) << 1) + p;
    int col  = cg * 16 + (lane & 15);
    SwB[o] = B[(size_t)row * Nc + col];
}

// ---------------------------------------------------------------------------
// fp32 WMMA GEMM:  C(MxNc) = A(MxK) * SwB [+ bias]
// One wave computes a 16x64 tile with V_WMMA_F32_16X16X4_F32.
// Requires M % 16 == 0, K % 4 == 0, Nc % 64 == 0 (M=50000=16*3125).
// A 16x4 fp32 layout: lanes 0-15 -> K={0,1}, lanes 16-31 -> K={2,3}, M = lane&15.
// B comes pre-swizzled (see swizzle_b).
// C/D: VGPR r holds row M = r + (lane<16?0:8), col N = lane&15.
// ---------------------------------------------------------------------------
__global__ void gemm_f32_wmma(const float* __restrict__ A,
                              const float* __restrict__ SwB,
                              float* __restrict__ C,
                              const float* __restrict__ bias,
                              int M, int K, int Nc) {
    const int wave = (blockIdx.x * blockDim.x + threadIdx.x) >> 5;
    const int lane = threadIdx.x & 31;
    const int ntn  = Nc >> 6;                 // 64-wide column groups
    const int tileM = wave / ntn;
    const int tileN = wave % ntn;
    if (tileM * 16 >= M) return;              // wave-uniform exit (EXEC all-1s inside)

    const int mrow   = tileM * 16 + (lane & 15);
    const int khalf  = (lane >> 4) << 1;      // 0 or 2
    const int cbase  = tileN * 64 + (lane & 15);
    const int kbstep = (Nc >> 4) * 64;        // SwB floats per k-block

    v8f acc0 = {}, acc1 = {}, acc2 = {}, acc3 = {};

    const float* arow = A + (size_t)mrow * K + khalf;           // 8B-aligned
    const float* bptr = SwB + ((size_t)tileN * 4 * 32 + lane) * 2;  // 8B-aligned
    for (int k0 = 0; k0 < K; k0 += 4) {
        v2f a  = *(const v2f*)(arow + k0);
        v2f b0 = *(const v2f*)(bptr);
        v2f b1 = *(const v2f*)(bptr + 64);
        v2f b2 = *(const v2f*)(bptr + 128);
        v2f b3 = *(const v2f*)(bptr + 192);
        acc0 = __builtin_amdgcn_wmma_f32_16x16x4_f32(false, a, false, b0, (short)0, acc0, false, false);
        acc1 = __builtin_amdgcn_wmma_f32_16x16x4_f32(false, a, false, b1, (short)0, acc1, false, false);
        acc2 = __builtin_amdgcn_wmma_f32_16x16x4_f32(false, a, false, b2, (short)0, acc2, false, false);
        acc3 = __builtin_amdgcn_wmma_f32_16x16x4_f32(false, a, false, b3, (short)0, acc3, false, false);
        bptr += kbstep;
    }

    float bv0 = 0.f, bv1 = 0.f, bv2 = 0.f, bv3 = 0.f;
    if (bias) {
        bv0 = bias[cbase];
        bv1 = bias[cbase + 16];
        bv2 = bias[cbase + 32];
        bv3 = bias[cbase + 48];
    }
    const int rowoff = (lane >> 4) << 3;      // 0 or 8
#pragma unroll
    for (int r = 0; r < 8; ++r) {
        size_t off = (size_t)(tileM * 16 + rowoff + r) * Nc + cbase;
        C[off]      = acc0[r] + bv0;
        C[off + 16] = acc1[r] + bv1;
        C[off + 32] = acc2[r] + bv2;
        C[off + 48] = acc3[r] + bv3;
    }
}

// ---------------------------------------------------------------------------
// Walk init: cur = identity, walk_idx[0] = identity
// ---------------------------------------------------------------------------
__global__ void init_walk(int* __restrict__ cur, int* __restrict__ walk0, int n) {
    int tid = blockIdx.x * blockDim.x + threadIdx.x;
    if (tid < n) { cur[tid] = tid; walk0[tid] = tid; }
}

// h_pre = OutBig[:, 0:64] + b1
__global__ void init_hpre(const float* __restrict__ OutBig,
                          const float* __restrict__ b1,
                          float* __restrict__ h_pre, int n) {
    int tid = blockIdx.x * blockDim.x + threadIdx.x;
    if (tid >= n * HMLP) return;
    int node = tid >> 6;
    int c    = tid & 63;
    h_pre[tid] = OutBig[(size_t)node * WBC + c] + b1[c];
}

// ---------------------------------------------------------------------------
// One walk step (wave per node).
//   logp[d] = sum_h relu(h_pre[n,h] + A_t[nbr_d,h]) * W2[h]
//   p = softmax_16(logp) + noise[t,n,:]; sel = first-argmax(p)
//   cur[n] = nbr[sel]; h_pre[n] += A_t[nbr[sel]]
// A_t[v] lives at OutBig[v, 64 + 64*t : 128 + 64*t] (L2-resident, 141MB < 192MB L2).
// ---------------------------------------------------------------------------
__global__ void walk_step(const float* __restrict__ OutBig,
                          float* __restrict__ h_pre,
                          const int* __restrict__ dst,      // N*DEG neighbor table
                          const float* __restrict__ noise_t,// N x 16 slice
                          const float* __restrict__ W2,     // 64
                          int* __restrict__ cur,
                          int* __restrict__ walk_next,
                          int t, int n_nodes) {
    const int wave = (blockIdx.x * blockDim.x + threadIdx.x) >> 5;
    const int lane = threadIdx.x & 31;
    if (wave >= n_nodes) return;
    const int n    = wave;
    const int d    = lane >> 1;       // neighbor 0..15
    const int half = lane & 1;        // h-range half

    const int c   = cur[n];
    const int nbr = dst[c * DEG + d];

    const float* arow = OutBig + (size_t)nbr * WBC + 64 + t * 64 + half * 32;
    const float* hrow = h_pre + (size_t)n * HMLP + half * 32;
    const float* wrow = W2 + half * 32;

    float part = 0.f;
#pragma unroll
    for (int h = 0; h < 32; ++h) {
        float v = hrow[h] + arow[h];
        part = fmaf(fmaxf(v, 0.f), wrow[h], part);
    }
    part += __shfl_xor(part, 1, 32);                 // pair-reduce halves
    float lp = __shfl(part, (lane & 15) << 1, 32);   // lane i holds logp[i&15]

    // softmax over 16 (butterflies stay within each 16-lane half)
    float m = lp;
    for (int off = 8; off; off >>= 1) m = fmaxf(m, __shfl_xor(m, off, 32));
    float e = expf(lp - m);
    float s = e;
    for (int off = 8; off; off >>= 1) s += __shfl_xor(s, off, 32);
    float p = e / s + noise_t[(size_t)n * DEG + (lane & 15)];

    // argmax, first occurrence on ties
    float bv = p;
    int   bi = lane & 15;
    for (int off = 8; off; off >>= 1) {
        float ov = __shfl_xor(bv, off, 32);
        int   oi = __shfl_xor(bi, off, 32);
        if (ov > bv || (ov == bv && oi < bi)) { bv = ov; bi = oi; }
    }

    int sel_nbr = __shfl(nbr, bi << 1, 32);          // broadcast chosen neighbor
    if (lane == 0) { cur[n] = sel_nbr; walk_next[n] = sel_nbr; }

    // h_pre[n] += A_t[sel_nbr]   (64 floats, 2 per lane)
    const float* aupd = OutBig + (size_t)sel_nbr * WBC + 64 + t * 64;
    float* hupd = h_pre + (size_t)n * HMLP;
    hupd[lane * 2]     += aupd[lane * 2];
    hupd[lane * 2 + 1] += aupd[lane * 2 + 1];
}

// ---------------------------------------------------------------------------
// GRU gate update. GX rows are gathered from OutBig[:,320:704] via walk index.
//   r = sig(gx_r + gh_r); z = sig(gx_z + gh_z); nh = tanh(gx_n + r*gh_n)
//   h = (1-z)*nh + z*h        (biases folded in here)
// 32 threads per node, 4 h-values per thread.
// ---------------------------------------------------------------------------
__global__ void gru_update(const float* __restrict__ OutBig,
                           const float* __restrict__ GH,    // N x 384  (h @ Wh)
                           const int* __restrict__ xidx,    // N walk indices for this step
                           const float* __restrict__ bx,    // 384
                           const float* __restrict__ bh,    // 384
                           float* __restrict__ h,           // N x 128
                           int n_nodes) {
    int tid = blockIdx.x * blockDim.x + threadIdx.x;
    int n = tid >> 5;
    if (n >= n_nodes) return;
    int l = tid & 31;
    const float* gx = OutBig + (size_t)xidx[n] * WBC + 320;
    const float* gh = GH + (size_t)n * 384;
    float* hp = h + (size_t)n * HGRU;
#pragma unroll
    for (int j = 0; j < 4; ++j) {
        int hh = l * 4 + j;
        float xr = gx[hh]       + bx[hh];
        float xz = gx[128 + hh] + bx[128 + hh];
        float xn = gx[256 + hh] + bx[256 + hh];
        float hr = gh[hh]       + bh[hh];
        float hz = gh[128 + hh] + bh[128 + hh];
        float hn = gh[256 + hh] + bh[256 + hh];
        float r  = 1.f / (1.f + expf(-(xr + hr)));
        float z  = 1.f / (1.f + expf(-(xz + hz)));
        float nh = tanhf(xn + r * hn);
        hp[hh] = (1.f - z) * nh + z * hp[hh];
    }
}

// ---------------------------------------------------------------------------
// Host launch
// ---------------------------------------------------------------------------
extern "C" void kernel_launch(void* const* d_in, const int* in_sizes, int n_in,
                              void* d_out, int out_size, void* d_ws, size_t ws_size,
                              hipStream_t stream) {
    const float* node_attr = (const float*)d_in[0];
    const int*   edge_idx  = (const int*)d_in[1];
    // d_in[2] = slices (implied by cur*DEG, unused)
    const float* noise     = (const float*)d_in[3];
    const float* W1        = (const float*)d_in[4];
    const float* b1        = (const float*)d_in[5];
    const float* W2        = (const float*)d_in[6];
    // d_in[7] = b2 (softmax-invariant, unused)
    const float* Wx        = (const float*)d_in[8];
    const float* Wh        = (const float*)d_in[9];
    const float* bx        = (const float*)d_in[10];
    const float* bh        = (const float*)d_in[11];
    const float* Wo        = (const float*)d_in[12];
    const float* bo        = (const float*)d_in[13];

    const int*   dst = edge_idx + NNODES * DEG;   // edge_index[1]

    // workspace partition (256-byte aligned slabs)
    char* ws = (char*)d_ws;
    size_t off = 0;
    auto alloc = [&](size_t bytes) {
        char* p = ws + off;
        off += (bytes + 255) & ~(size_t)255;
        return p;
    };
    float* Wbig   = (float*)alloc((size_t)CIN * WBC * sizeof(float));
    float* SwWbig = (float*)alloc((size_t)CIN * WBC * sizeof(float));
    float* SwWh   = (float*)alloc((size_t)CIN * 384 * sizeof(float));
    float* SwWo   = (float*)alloc((size_t)CIN * 128 * sizeof(float));
    float* OutBig = (float*)alloc((size_t)NNODES * WBC * sizeof(float));
    float* h_pre  = (float*)alloc((size_t)NNODES * HMLP * sizeof(float));
    float* GH     = (float*)alloc((size_t)NNODES * 384 * sizeof(float));
    float* hbuf   = (float*)alloc((size_t)NNODES * HGRU * sizeof(float));
    int*   cur    = (int*)alloc((size_t)NNODES * sizeof(int));
    int*   widx   = (int*)alloc((size_t)(TSTEPS + 1) * NNODES * sizeof(int));
    (void)ws_size; (void)in_sizes; (void)n_in; (void)out_size;

    // 1) weight prep: Wbig assembly + swizzles into WMMA B-fragment layout
    build_wbig<<<(CIN * WBC + 255) / 256, 256, 0, stream>>>(W1, Wx, Wbig);
    swizzle_b<<<(CIN * WBC + 255) / 256, 256, 0, stream>>>(Wbig, SwWbig, CIN, WBC);
    swizzle_b<<<(CIN * 384 + 255) / 256, 256, 0, stream>>>(Wh, SwWh, CIN, 384);
    swizzle_b<<<(CIN * 128 + 255) / 256, 256, 0, stream>>>(Wo, SwWo, CIN, 128);

    // 2) fused projection GEMM: OutBig = node_attr @ Wbig
    {
        int waves = (NNODES / 16) * (WBC / 64);   // 3125 * 11
        gemm_f32_wmma<<<(waves + 7) / 8, 256, 0, stream>>>(
            node_attr, SwWbig, OutBig, nullptr, NNODES, CIN, WBC);
    }

    // 3) walk init
    init_walk<<<(NNODES + 255) / 256, 256, 0, stream>>>(cur, widx, NNODES);
    init_hpre<<<(NNODES * HMLP + 255) / 256, 256, 0, stream>>>(OutBig, b1, h_pre, NNODES);

    // 4) walk steps (sequential)
    for (int t = 0; t < TSTEPS; ++t) {
        walk_step<<<(NNODES * 32 + 255) / 256, 256, 0, stream>>>(
            OutBig, h_pre, dst, noise + (size_t)t * NNODES * DEG, W2,
            cur, widx + (size_t)(t + 1) * NNODES, t, NNODES);
    }

    // 5) GRU over 5 time steps; h0 = 0
    hipMemsetAsync(hbuf, 0, (size_t)NNODES * HGRU * sizeof(float), stream);
    for (int t = 0; t <= TSTEPS; ++t) {
        int waves = (NNODES / 16) * (384 / 64);   // 3125 * 6
        gemm_f32_wmma<<<(waves + 7) / 8, 256, 0, stream>>>(
            hbuf, SwWh, GH, nullptr, NNODES, HGRU, 384);
        gru_update<<<(NNODES * 32 + 255) / 256, 256, 0, stream>>>(
            OutBig, GH, widx + (size_t)t * NNODES, bx, bh, hbuf, NNODES);
    }

    // 6) output projection: out = h @ Wo + bo
    {
        int waves = (NNODES / 16) * (CIN / 64);   // 3125 * 2
        gemm_f32_wmma<<<(waves + 7) / 8, 256, 0, stream>>>(
            hbuf, SwWo, (float*)d_out, bo, NNODES, HGRU, CIN);
    }
}